// Slater_2765958939212
// MI455X (gfx1250) — compile-verified
//
#include <hip/hip_runtime.h>
#include <math.h>

#define N_ORB  128
#define NOCC   64
#define N_DETS 16
#define BATCH  1024
#define LDSTR  68   // LDS row stride in floats (64 + pad, avoids bank conflicts)

typedef float v2f __attribute__((ext_vector_type(2)));
typedef float v8f __attribute__((ext_vector_type(8)));

// ---------------------------------------------------------------------------
// Kernel 1: one wave (32 lanes) per (batch, det). Gathers the alpha and beta
// 64x64 complex submatrices into LDS, does blocked LU (NB=16) with partial
// pivoting, trailing updates via V_WMMA_F32_16X16X4_F32 (complex = 4 real
// GEMMs, A negated in VALU since f32 WMMA has no A/B NEG modifier).
// Writes complex logdet_alpha + logdet_beta to ws[(b*16+d)*2 .. +1].
// ---------------------------------------------------------------------------
__global__ __launch_bounds__(32) void slater_lu_kernel(
    const int*   __restrict__ s,     // (BATCH, 256) int32 occupancies
    const float* __restrict__ M,     // (16, 128, 64) complex64 -> float pairs
    float*       __restrict__ ws)    // (BATCH*16) complex64 scratch
{
    const int wg   = blockIdx.x;     // 0 .. BATCH*N_DETS-1
    const int b    = wg >> 4;
    const int d    = wg & 15;
    const int lane = threadIdx.x;    // 0..31 (wave32)

    __shared__ float Ar[NOCC * LDSTR];
    __shared__ float Ai[NOCC * LDSTR];
    __shared__ int   rows[NOCC];

    float tot_re = 0.0f, tot_im = 0.0f;

    for (int sp = 0; sp < 2; ++sp) {
        // ---- occupied row list (64 ones guaranteed) ----
        if (lane == 0) {
            const int* sv = s + b * (2 * N_ORB) + sp * N_ORB;
            int cnt = 0;
            for (int k = 0; k < N_ORB; ++k)
                if (sv[k] != 0 && cnt < NOCC) rows[cnt++] = k;
            for (; cnt < NOCC; ++cnt) rows[cnt] = 0;
        }
        __builtin_amdgcn_wave_barrier();   // same-wave LDS ops are in-order;
                                           // this just blocks compiler motion
        // ---- gather A = M[d, rows, :64] into LDS (re/im planes) ----
        const float* Md = M + (size_t)d * N_ORB * NOCC * 2;
        for (int idx = lane; idx < NOCC * NOCC; idx += 32) {
            int r = idx >> 6, c = idx & 63;
            int o = rows[r];
            Ar[r * LDSTR + c] = Md[(o * NOCC + c) * 2 + 0];
            Ai[r * LDSTR + c] = Md[(o * NOCC + c) * 2 + 1];
        }
        __builtin_amdgcn_wave_barrier();

        float sumLog = 0.0f, sumArg = 0.0f;
        int   nswaps = 0;

        for (int k0 = 0; k0 < NOCC; k0 += 16) {
            const int kend = k0 + 16;

            // ---- panel factorization (cols k0..kend-1), partial pivoting ----
            for (int j = k0; j < kend; ++j) {
                // argmax |A[r][j]|^2 over rows r = j..63 (lane owns r, r+32)
                float bestv = -1.0f; int besti = j;
                {
                    int r1 = lane;
                    if (r1 >= j) {
                        float xr = Ar[r1 * LDSTR + j], xi = Ai[r1 * LDSTR + j];
                        float m2 = xr * xr + xi * xi;
                        if (m2 > bestv) { bestv = m2; besti = r1; }
                    }
                    int r2 = lane + 32;
                    {
                        float xr = Ar[r2 * LDSTR + j], xi = Ai[r2 * LDSTR + j];
                        float m2 = xr * xr + xi * xi;
                        if (r2 >= j && m2 > bestv) { bestv = m2; besti = r2; }
                    }
                }
                for (int off = 16; off > 0; off >>= 1) {
                    float ov = __shfl_xor(bestv, off, 32);
                    int   oi = __shfl_xor(besti, off, 32);
                    if (ov > bestv) { bestv = ov; besti = oi; }
                }
                const int p = besti;
                if (p != j) {
                    nswaps ^= 1;
                    for (int c = lane; c < NOCC; c += 32) {
                        float t;
                        t = Ar[j * LDSTR + c]; Ar[j * LDSTR + c] = Ar[p * LDSTR + c]; Ar[p * LDSTR + c] = t;
                        t = Ai[j * LDSTR + c]; Ai[j * LDSTR + c] = Ai[p * LDSTR + c]; Ai[p * LDSTR + c] = t;
                    }
                }
                __builtin_amdgcn_wave_barrier();

                const float ur = Ar[j * LDSTR + j], ui = Ai[j * LDSTR + j];
                const float m2 = ur * ur + ui * ui;
                sumLog += 0.5f * __logf(m2);
                sumArg += atan2f(ui, ur);
                const float inv = 1.0f / m2;

                // multipliers + rank-1 update restricted to panel columns
                for (int rr = 0; rr < 2; ++rr) {
                    const int i = lane + rr * 32;
                    if (i > j) {
                        float arv = Ar[i * LDSTR + j], aiv = Ai[i * LDSTR + j];
                        float mr = (arv * ur + aiv * ui) * inv;
                        float mi = (aiv * ur - arv * ui) * inv;
                        Ar[i * LDSTR + j] = mr; Ai[i * LDSTR + j] = mi;
                        for (int jj = j + 1; jj < kend; ++jj) {
                            float pr = Ar[j * LDSTR + jj], pi = Ai[j * LDSTR + jj];
                            Ar[i * LDSTR + jj] -= mr * pr - mi * pi;
                            Ai[i * LDSTR + jj] -= mr * pi + mi * pr;
                        }
                    }
                }
                __builtin_amdgcn_wave_barrier();
            }

            if (kend < NOCC) {
                // ---- triangular solve: L11 * U12 = A12 (rows k0..kend-1) ----
                for (int i = k0 + 1; i < kend; ++i) {
                    for (int c = kend + lane; c < NOCC; c += 32) {
                        float accr = Ar[i * LDSTR + c], acci = Ai[i * LDSTR + c];
                        for (int t = k0; t < i; ++t) {
                            float lr = Ar[i * LDSTR + t], li = Ai[i * LDSTR + t];
                            float xr = Ar[t * LDSTR + c], xi = Ai[t * LDSTR + c];
                            accr -= lr * xr - li * xi;
                            acci -= lr * xi + li * xr;
                        }
                        Ar[i * LDSTR + c] = accr; Ai[i * LDSTR + c] = acci;
                    }
                }
                __builtin_amdgcn_wave_barrier();

                // ---- trailing update A22 -= L21 * U12 via f32 WMMA ----
                const int m  = NOCC - kend;   // 48, 32, 16
                const int nt = m >> 4;
                const int cm = (lane >> 4) * 8;   // C: M = v + cm
                const int cn = lane & 15;         // C: N = cn
                const int am = lane & 15;         // A: M = am
                const int ak = (lane >> 4) * 2;   // A/B: K pair base

                for (int ti = 0; ti < nt; ++ti) {
                    for (int tj = 0; tj < nt; ++tj) {
                        const int rowB = kend + ti * 16;
                        const int colB = kend + tj * 16;
                        v8f cr, ci;
                        for (int v = 0; v < 8; ++v) {
                            cr[v] = Ar[(rowB + v + cm) * LDSTR + colB + cn];
                            ci[v] = Ai[(rowB + v + cm) * LDSTR + colB + cn];
                        }
                        for (int ks = 0; ks < 4; ++ks) {
                            const int kb = k0 + ks * 4;
                            v2f arf, aif, brf, bif, nar, nai;
                            arf[0] = Ar[(rowB + am) * LDSTR + kb + ak];
                            arf[1] = Ar[(rowB + am) * LDSTR + kb + ak + 1];
                            aif[0] = Ai[(rowB + am) * LDSTR + kb + ak];
                            aif[1] = Ai[(rowB + am) * LDSTR + kb + ak + 1];
                            brf[0] = Ar[(kb + ak)     * LDSTR + colB + cn];
                            brf[1] = Ar[(kb + ak + 1) * LDSTR + colB + cn];
                            bif[0] = Ai[(kb + ak)     * LDSTR + colB + cn];
                            bif[1] = Ai[(kb + ak + 1) * LDSTR + colB + cn];
                            nar[0] = -arf[0]; nar[1] = -arf[1];
                            nai[0] = -aif[0]; nai[1] = -aif[1];
                            // Cr += -Ar*Br + Ai*Bi ; Ci += -Ar*Bi - Ai*Br
                            cr = __builtin_amdgcn_wmma_f32_16x16x4_f32(
                                     false, nar, false, brf, (short)0, cr, false, false);
                            cr = __builtin_amdgcn_wmma_f32_16x16x4_f32(
                                     false, aif, false, bif, (short)0, cr, false, false);
                            ci = __builtin_amdgcn_wmma_f32_16x16x4_f32(
                                     false, nar, false, bif, (short)0, ci, false, false);
                            ci = __builtin_amdgcn_wmma_f32_16x16x4_f32(
                                     false, nai, false, brf, (short)0, ci, false, false);
                        }
                        for (int v = 0; v < 8; ++v) {
                            Ar[(rowB + v + cm) * LDSTR + colB + cn] = cr[v];
                            Ai[(rowB + v + cm) * LDSTR + colB + cn] = ci[v];
                        }
                    }
                }
                __builtin_amdgcn_wave_barrier();
            }
        }

        tot_re += sumLog;
        tot_im += sumArg + (nswaps ? 3.14159265358979323846f : 0.0f);
        __builtin_amdgcn_wave_barrier();
    }

    if (lane == 0) {
        ws[(b * N_DETS + d) * 2 + 0] = tot_re;
        ws[(b * N_DETS + d) * 2 + 1] = tot_im;
    }
}

// ---------------------------------------------------------------------------
// Kernel 2: complex logsumexp over the 16 determinants per batch element.
// ---------------------------------------------------------------------------
__global__ __launch_bounds__(256) void slater_lse_kernel(
    const float* __restrict__ ws,
    const float* __restrict__ logc,   // (16,) complex64 -> float pairs
    float*       __restrict__ out)    // (BATCH,) complex64 -> float pairs
{
    const int b = blockIdx.x * blockDim.x + threadIdx.x;
    if (b >= BATCH) return;

    float xr[N_DETS], xi[N_DETS];
    float m = -3.0e38f;
    for (int d = 0; d < N_DETS; ++d) {
        xr[d] = ws[(b * N_DETS + d) * 2 + 0] + logc[2 * d + 0];
        xi[d] = ws[(b * N_DETS + d) * 2 + 1] + logc[2 * d + 1];
        m = fmaxf(m, xr[d]);
    }
    float sr = 0.0f, si = 0.0f;
    for (int d = 0; d < N_DETS; ++d) {
        float e = __expf(xr[d] - m);
        sr += e * __cosf(xi[d]);
        si += e * __sinf(xi[d]);
    }
    out[2 * b + 0] = 0.5f * __logf(sr * sr + si * si) + m;
    out[2 * b + 1] = atan2f(si, sr);
}

extern "C" void kernel_launch(void* const* d_in, const int* in_sizes, int n_in,
                              void* d_out, int out_size, void* d_ws, size_t ws_size,
                              hipStream_t stream)
{
    const int*   s    = (const int*)d_in[0];    // (1024, 256) int32
    const float* M    = (const float*)d_in[1];  // (16, 128, 64) complex64
    const float* logc = (const float*)d_in[2];  // (16,) complex64
    float*       out  = (float*)d_out;
    float*       ws   = (float*)d_ws;           // needs 16384*2*4 = 128 KB

    slater_lu_kernel<<<BATCH * N_DETS, 32, 0, stream>>>(s, M, ws);
    slater_lse_kernel<<<(BATCH + 255) / 256, 256, 0, stream>>>(ws, logc, out);
}